// ReformerAttention_61435212202310
// MI455X (gfx1250) — compile-verified
//
#include <hip/hip_runtime.h>

typedef __attribute__((ext_vector_type(2))) float v2f;
typedef __attribute__((ext_vector_type(8))) float v8f;

#define WMMA_F32_4(a, b, c) \
    __builtin_amdgcn_wmma_f32_16x16x4_f32(false, (a), false, (b), (short)0, (c), false, false)

// ---------------------------------------------------------------------------
// Kernel 0: zero argmax slots, build zero-padded lsh (16 x D) in workspace
// ---------------------------------------------------------------------------
__global__ void rfa_init_kernel(const float* __restrict__ lsh,
                                float* __restrict__ lsh_pad,
                                unsigned long long* __restrict__ slots,
                                int H, int D) {
    int idx = blockIdx.x * blockDim.x + threadIdx.x;
    int total = 16 * D;
    if (idx < total) {
        int row = idx / D;
        lsh_pad[idx] = (row < H) ? lsh[idx] : 0.0f;  // flat index matches for row < H
    }
    if (idx < 32) slots[idx] = 0ull;                 // B*H = 32 argmax slots
}

// ---------------------------------------------------------------------------
// Kernel 1: k_proj via f32 WMMA + global argmax over n per (b,h)
// grid = (N/16, B), block = 32 (one wave per 16-row tile)
// ---------------------------------------------------------------------------
__global__ __launch_bounds__(32)
void rfa_kproj_argmax_kernel(const float* __restrict__ k,
                             const float* __restrict__ lsh_pad,
                             unsigned long long* __restrict__ slots,
                             int N, int D) {
    const int lane = threadIdx.x;
    const int b    = blockIdx.y;
    const int row0 = blockIdx.x * 16;
    const int half = lane >> 4;     // 0 or 1
    const int l16  = lane & 15;

    const float* krow = k       + ((size_t)b * N + row0 + l16) * D;
    const float* lrow = lsh_pad + (size_t)l16 * D;

    v8f c = {};
    #pragma unroll 8
    for (int k0 = 0; k0 < D; k0 += 4) {
        const int off = k0 + 2 * half;
        v2f a  = *(const v2f*)(krow + off);   // A: k tile, M=l16, K pair
        v2f bb = *(const v2f*)(lrow + off);   // B: lsh^T,  N=l16, K pair
        c = WMMA_F32_4(a, bb, c);
    }
    // D-layout: lane holds kproj[m = r + 8*half][h = l16] in c[r]
    float bv = c[0];
    int   bm = 8 * half;
    #pragma unroll
    for (int r = 1; r < 8; ++r) {
        float vv = c[r];
        if (vv > bv) { bv = vv; bm = r + 8 * half; }   // strict > keeps smallest m
    }
    // combine halves (other half has strictly larger/smaller m set; tie -> min m)
    float ov = __shfl_xor(bv, 16, 32);
    int   om = __shfl_xor(bm, 16, 32);
    if (ov > bv || (ov == bv && om < bm)) { bv = ov; bm = om; }

    if (half == 0 && l16 < 8) {
        // monotone key for f32, inverted index so ties pick smallest n under max
        unsigned u = __float_as_uint(bv);
        u = (u & 0x80000000u) ? ~u : (u | 0x80000000u);
        unsigned n = (unsigned)(row0 + bm);
        unsigned long long packed =
            ((unsigned long long)u << 32) | (unsigned long long)(0xFFFFFFFFu - n);
        atomicMax(&slots[b * 8 + l16], packed);
    }
}

// ---------------------------------------------------------------------------
// Kernel 2: stable argsort of 8 buckets, gather permuted rows 0..7 of k/v
// into zero-padded 16 x D tiles.  grid = (16, B), block = 256
// ---------------------------------------------------------------------------
__global__ void rfa_gather_kernel(const float* __restrict__ k,
                                  const float* __restrict__ v,
                                  const unsigned long long* __restrict__ slots,
                                  float* __restrict__ k_pad,
                                  float* __restrict__ v_pad,
                                  int N, int D) {
    const int j = blockIdx.x;   // destination row 0..15
    const int b = blockIdx.y;
    float* kd = k_pad + ((size_t)b * 16 + j) * D;
    float* vd = v_pad + ((size_t)b * 16 + j) * D;

    if (j >= 8) {   // zero padding rows
        for (int t = threadIdx.x; t < D; t += blockDim.x) { kd[t] = 0.0f; vd[t] = 0.0f; }
        return;
    }
    unsigned bucket[8];
    #pragma unroll
    for (int h = 0; h < 8; ++h)
        bucket[h] = 0xFFFFFFFFu - (unsigned)(slots[b * 8 + h] & 0xFFFFFFFFull);
    // stable argsort: src = h whose rank == j
    int src = 0;
    #pragma unroll
    for (int h = 0; h < 8; ++h) {
        int rank = 0;
        #pragma unroll
        for (int h2 = 0; h2 < 8; ++h2)
            rank += (bucket[h2] < bucket[h]) || (bucket[h2] == bucket[h] && h2 < h);
        if (rank == j) src = h;
    }
    // NOTE: reference gathers k/v at the argsort indices themselves (values in [0,8))
    const float* ks = k + ((size_t)b * N + src) * D;
    const float* vs = v + ((size_t)b * N + src) * D;
    for (int t = threadIdx.x; t < D; t += blockDim.x) { kd[t] = ks[t]; vd[t] = vs[t]; }
}

// ---------------------------------------------------------------------------
// Kernel 3: sims = q . k_pad^T (WMMA), masked softmax over 8, out = attn . v_pad
// grid = (N/16, B), block = 32 (single wave; __syncthreads is S_NOP but orders LDS)
// ---------------------------------------------------------------------------
__global__ __launch_bounds__(32)
void rfa_attn_kernel(const float* __restrict__ q,
                     const float* __restrict__ k_pad,
                     const float* __restrict__ v_pad,
                     float* __restrict__ out,
                     int N, int D) {
    __shared__ float attn_lds[16 * 16];

    const int lane = threadIdx.x;
    const int b    = blockIdx.y;
    const int row0 = blockIdx.x * 16;
    const int half = lane >> 4;
    const int l16  = lane & 15;

    const float* qrow = q     + ((size_t)b * N + row0 + l16) * D;
    const float* krow = k_pad + ((size_t)b * 16 + l16) * D;

    // --- GEMM 1: sims[16 x 16] = q_tile[16 x D] * k_pad^T ---
    v8f c = {};
    #pragma unroll 8
    for (int k0 = 0; k0 < D; k0 += 4) {
        const int off = k0 + 2 * half;
        v2f a  = *(const v2f*)(qrow + off);
        v2f bb = *(const v2f*)(krow + off);
        c = WMMA_F32_4(a, bb, c);
    }

    // --- softmax over the 8 valid hash columns (lane column = l16) ---
    const bool valid = (l16 < 8);
    float attn_r[8];
    #pragma unroll
    for (int r = 0; r < 8; ++r) {
        float s = valid ? c[r] : -3.0e38f;
        float m = s;
        m = fmaxf(m, __shfl_xor(m, 1, 32));
        m = fmaxf(m, __shfl_xor(m, 2, 32));
        m = fmaxf(m, __shfl_xor(m, 4, 32));
        float e = valid ? __expf(c[r] - m) : 0.0f;
        float sum = e;
        sum += __shfl_xor(sum, 1, 32);
        sum += __shfl_xor(sum, 2, 32);
        sum += __shfl_xor(sum, 4, 32);
        attn_r[r] = valid ? (e / sum) : 0.0f;
    }

    // --- reshape D-layout -> A-layout through LDS ---
    #pragma unroll
    for (int r = 0; r < 8; ++r)
        attn_lds[(r + 8 * half) * 16 + l16] = attn_r[r];
    __syncthreads();

    v2f afrag[4];
    #pragma unroll
    for (int ks = 0; ks < 4; ++ks) {
        const int kk = 4 * ks + 2 * half;
        afrag[ks][0] = attn_lds[l16 * 16 + kk];
        afrag[ks][1] = attn_lds[l16 * 16 + kk + 1];
    }

    // --- GEMM 2: out_tile[16 x D] = attn[16 x 16] * v_pad[16 x D] ---
    const float* vbase = v_pad + (size_t)b * 16 * D;
    float* obase = out + ((size_t)b * N + row0) * D;
    for (int ct = 0; ct < D; ct += 16) {
        v8f acc = {};
        #pragma unroll
        for (int ks = 0; ks < 4; ++ks) {
            const int kk = 4 * ks + 2 * half;
            v2f bb;
            bb[0] = vbase[(size_t)kk * D + ct + l16];
            bb[1] = vbase[(size_t)(kk + 1) * D + ct + l16];
            acc = WMMA_F32_4(afrag[ks], bb, acc);
        }
        #pragma unroll
        for (int r = 0; r < 8; ++r)
            obase[(size_t)(r + 8 * half) * D + ct + l16] = acc[r];
    }
}

// ---------------------------------------------------------------------------
extern "C" void kernel_launch(void* const* d_in, const int* in_sizes, int n_in,
                              void* d_out, int out_size, void* d_ws, size_t ws_size,
                              hipStream_t stream) {
    const float* q   = (const float*)d_in[0];
    const float* k   = (const float*)d_in[1];
    const float* v   = (const float*)d_in[2];
    const float* lsh = (const float*)d_in[3];
    float* out = (float*)d_out;

    const int B = 4, N = 8192, D = 1024, H = 8;

    // workspace layout
    char* ws = (char*)d_ws;
    unsigned long long* slots = (unsigned long long*)ws;          // 32 * 8B
    float* lsh_pad = (float*)(ws + 256);                          // 16*D floats
    float* k_pad   = (float*)(ws + 256 + (size_t)16 * D * 4);     // B*16*D floats
    float* v_pad   = (float*)(ws + 256 + (size_t)16 * D * 4 + (size_t)B * 16 * D * 4);

    rfa_init_kernel<<<(16 * D + 255) / 256, 256, 0, stream>>>(lsh, lsh_pad, slots, H, D);

    dim3 g1(N / 16, B);
    rfa_kproj_argmax_kernel<<<g1, 32, 0, stream>>>(k, lsh_pad, slots, N, D);

    dim3 g2(16, B);
    rfa_gather_kernel<<<g2, 256, 0, stream>>>(k, v, slots, k_pad, v_pad, N, D);

    dim3 g3(N / 16, B);
    rfa_attn_kernel<<<g3, 32, 0, stream>>>(q, k_pad, v_pad, out, N, D);
}